// DecoderModel_80848464379939
// MI455X (gfx1250) — compile-verified
//
#include <hip/hip_runtime.h>
#include <hip/hip_bf16.h>

// Problem constants (match reference)
#define BATCH 128
#define NNODE 1024
#define UDIM  128
#define CIN   129             // OD + U
#define NMAT  3               // K+1
#define CPAD  144             // CIN padded to multiple of 16 (9 WMMA col tiles)
#define ROWF  (NMAT * CPAD)   // 432 padded features per (b,n) row
#define NROWS (BATCH * NNODE) // 131072

// TDM-padded LDS strides (hardware pad_interval/pad_amount reproduces these):
//  S tile:  64 dw rows, pad 4 dw every 64 dw  -> stride 68   (banks: 4m mod 64 distinct)
//  X tile: 144 dw rows, pad 4 dw every 16 dw  -> stride 180  (channel c at c + (c>>4)*4)
//  A tile:  48 dw rows, pad 4 dw every 16 dw  -> stride 60
#define SS_STRIDE 68
#define SX_STRIDE 180
#define SA_STRIDE 60

typedef float v2f __attribute__((ext_vector_type(2)));
typedef float v8f __attribute__((ext_vector_type(8)));
typedef unsigned int u32x4 __attribute__((ext_vector_type(4)));
typedef int i32x4 __attribute__((ext_vector_type(4)));
typedef int i32x8 __attribute__((ext_vector_type(8)));

#if __has_include(<hip/amd_detail/amd_gfx1250_TDM.h>)
#define TDM_ARITY6 1
#else
#define TDM_ARITY6 0
#endif

__device__ __forceinline__ v8f wmma_f32(v2f a, v2f b, v8f c) {
    return __builtin_amdgcn_wmma_f32_16x16x4_f32(false, a, false, b, (short)0, c,
                                                 false, false);
}

// ---------------------------------------------------------------------------
// Tensor Data Mover: 2D tile load (global -> LDS), D# per CDNA5 ISA ch.8.
//   lds_off        : byte offset of destination in LDS
//   gaddr          : global address of tile start
//   tensor_d0      : tensor dim0 in dwords (OOB guard, >= tile_d0)
//   tile_d0        : dwords per row to copy
//   tile_d1        : number of rows
//   row_stride_dw  : global row stride in dwords
//   pad_int/pad_amt: D# pad codes (0/0 = no LDS padding)
// ---------------------------------------------------------------------------
__device__ __forceinline__ void tdm_load_2d(unsigned lds_off, const void* gaddr,
                                            unsigned tensor_d0, unsigned tile_d0,
                                            unsigned tile_d1, unsigned row_stride_dw,
                                            unsigned pad_int, unsigned pad_amt) {
    unsigned long long ga = (unsigned long long)(uintptr_t)gaddr;
    u32x4 g0;
    g0[0] = 1u;                                   // count=1, user descriptor
    g0[1] = lds_off;                              // LDS byte address
    g0[2] = (unsigned)(ga & 0xffffffffu);         // global_addr[31:0]
    g0[3] = (unsigned)((ga >> 32) & 0x01ffffffu)  // global_addr[56:32]
            | (2u << 30);                         // type = 2 ("image")
    unsigned d0 = (2u << 16);                     // data_size = 4 bytes
    if (pad_amt | pad_int)
        d0 |= (1u << 20) | (pad_int << 22) | (pad_amt << 25);  // pad_enable
    i32x8 g1;
    g1[0] = (int)d0;
    g1[1] = (int)((tensor_d0 & 0xffffu) << 16);                       // td0[15:0] @48
    g1[2] = (int)(((tensor_d0 >> 16) & 0xffffu) |
                  ((tile_d1 & 0xffffu) << 16));                       // td0 hi | td1 lo
    g1[3] = (int)(((tile_d1 >> 16) & 0xffffu) |
                  ((tile_d0 & 0xffffu) << 16));                       // td1 hi | tile_dim0
    g1[4] = (int)(tile_d1 & 0xffffu);                                 // tile_dim1
    g1[5] = (int)row_stride_dw;                                       // dim0_stride lo32
    g1[6] = 0;
    g1[7] = 0;
    i32x4 z4 = {0, 0, 0, 0};
#if TDM_ARITY6
    i32x8 z8 = {0, 0, 0, 0, 0, 0, 0, 0};
    __builtin_amdgcn_tensor_load_to_lds(g0, g1, z4, z4, z8, 0);
#else
    __builtin_amdgcn_tensor_load_to_lds(g0, g1, z4, z4, 0);
#endif
}

__device__ __forceinline__ unsigned lds_offset_of(const void* p) {
    // __shared__ pointers carry the LDS aperture in the high 32 bits;
    // low 32 bits are the LDS byte offset.
    return (unsigned)(uintptr_t)p;
}

// ---------------------------------------------------------------------------
// Permute + pad weights: W'[k*CPAD + c][j] = (c<129) ? W[c*3+k][j] : 0
// ---------------------------------------------------------------------------
__global__ __launch_bounds__(256) void prep_w_kernel(
    const float* __restrict__ gw, const float* __restrict__ cw,
    float* __restrict__ wpg, float* __restrict__ wpc) {
    int i = blockIdx.x * 256 + threadIdx.x;
    if (i < ROWF * 256) {
        int kk = i >> 8, j = i & 255;
        int k = kk / CPAD, c = kk % CPAD;
        wpg[i] = (c < CIN) ? gw[(c * NMAT + k) * 256 + j] : 0.0f;
    }
    if (i < ROWF * 128) {
        int kk = i >> 7, j = i & 127;
        int k = kk / CPAD, c = kk % CPAD;
        wpc[i] = (c < CIN) ? cw[(c * NMAT + k) * 128 + j] : 0.0f;
    }
}

// ---------------------------------------------------------------------------
// Build x0 = concat(inputs, h) into XK section 0 (channels padded with zeros)
// ---------------------------------------------------------------------------
__global__ __launch_bounds__(256) void build_x0_kernel(
    float* __restrict__ XK, const float* __restrict__ inp,
    const float* __restrict__ h) {
    size_t i = (size_t)blockIdx.x * 256 + threadIdx.x;  // over NROWS*CPAD
    if (i >= (size_t)NROWS * CPAD) return;
    int c = (int)(i % CPAD);
    size_t row = i / CPAD;
    float v;
    if (c == 0)           v = inp[row];
    else if (c <= UDIM)   v = h[row * UDIM + (c - 1)];
    else                  v = 0.0f;
    XK[row * ROWF + c] = v;
}

// ---------------------------------------------------------------------------
// Diffusion step: XK[:,out_sec] = alpha * (S @ XK[:,in_sec]) + beta*XK[:,sub_sec]
// Per block: one batch b, 64-node output strip, all 144 channels.
// Tiles staged by the Tensor Data Mover; 4 waves x 9 col-tiles of
// V_WMMA_F32_16X16X4_F32, K=1024.
// ---------------------------------------------------------------------------
__global__ __launch_bounds__(128) void diffusion_kernel(
    const float* __restrict__ S, float* __restrict__ XK,
    int in_sec, int out_sec, int sub_sec, float alpha, float beta) {
    __shared__ float sS[64 * SS_STRIDE];   // 64 x 64 S tile, TDM-padded
    __shared__ float sX[64 * SX_STRIDE];   // 64 x 144 X tile, TDM-padded

    int b   = blockIdx.x >> 4;             // N / 64 = 16 strips
    int m0  = (blockIdx.x & 15) * 64;
    int tid = threadIdx.x;
    int wave = tid >> 5;
    int lane = tid & 31;
    int lhalf = lane >> 4;
    int lmod  = lane & 15;

    const float* Xin = XK + (size_t)b * NNODE * ROWF + in_sec * CPAD;
    unsigned sS_off = lds_offset_of(&sS[0]);
    unsigned sX_off = lds_offset_of(&sX[0]);

    v8f acc[9];
#pragma unroll
    for (int t = 0; t < 9; ++t) acc[t] = (v8f)0.0f;

    for (int k0 = 0; k0 < NNODE; k0 += 64) {
        if (tid < 32) {
            // S tile: 64 rows x 64 dw, stride 1024; pad 4dw/64dw -> stride 68
            tdm_load_2d(sS_off, S + (size_t)m0 * NNODE + k0,
                        64, 64, 64, NNODE, /*pad_int=*/5, /*pad_amt=*/3);
            // X tile: 64 rows x 144 dw, stride 432; pad 4dw/16dw -> stride 180
            tdm_load_2d(sX_off, Xin + (size_t)k0 * ROWF,
                        CPAD, CPAD, 64, ROWF, /*pad_int=*/3, /*pad_amt=*/3);
            __builtin_amdgcn_s_wait_tensorcnt(0);
        }
        __syncthreads();

#pragma unroll
        for (int kk = 0; kk < 64; kk += 4) {
            int kr = kk + lhalf * 2;
            v2f a = *(const v2f*)&sS[(wave * 16 + lmod) * SS_STRIDE + kr];
#pragma unroll
            for (int ct = 0; ct < 9; ++ct) {
                v2f bf;
                bf.x = sX[(kr + 0) * SX_STRIDE + ct * 20 + lmod];  // c=ct*16 -> +ct*20
                bf.y = sX[(kr + 1) * SX_STRIDE + ct * 20 + lmod];
                acc[ct] = wmma_f32(a, bf, acc[ct]);
            }
        }
        __syncthreads();
    }

    float* Xout = XK + (size_t)b * NNODE * ROWF + out_sec * CPAD;
    const float* Xsub = XK + (size_t)b * NNODE * ROWF + sub_sec * CPAD;
#pragma unroll
    for (int ct = 0; ct < 9; ++ct) {
#pragma unroll
        for (int v = 0; v < 8; ++v) {
            int m = m0 + wave * 16 + lhalf * 8 + v;   // C-layout: VGPRv -> M=v / v+8
            int c = ct * 16 + lmod;
            float val = alpha * acc[ct][v];
            if (beta != 0.0f) val += beta * Xsub[(size_t)m * ROWF + c];
            Xout[(size_t)m * ROWF + c] = val;
        }
    }
}

// ---------------------------------------------------------------------------
// Gate GEMM: (64 rows x 256 cols) per block, K = 432 (padded features).
// TDM-staged tiles. Epilogue: sigmoid -> r*h written in place into XK x0
// slot, u to Ubuf. 8 waves: wave&3 = 16-row strip, wave>>2 = 128-col half.
// ---------------------------------------------------------------------------
__global__ __launch_bounds__(256) void gemm_gate_kernel(
    float* __restrict__ XK, const float* __restrict__ Wp,
    const float* __restrict__ gb, const float* __restrict__ h,
    float* __restrict__ Ubuf) {
    __shared__ float sA[64 * SA_STRIDE];   // 64 rows x 48 K-slice, TDM-padded
    __shared__ float sW[48 * 256];

    int row0 = blockIdx.x * 64;
    int tid = threadIdx.x;
    int wave = tid >> 5, lane = tid & 31;
    int lhalf = lane >> 4, lmod = lane & 15;
    int rstrip = (wave & 3) * 16;
    int chalf  = (wave >> 2) * 128;
    unsigned sA_off = lds_offset_of(&sA[0]);
    unsigned sW_off = lds_offset_of(&sW[0]);

    v8f acc[8];
#pragma unroll
    for (int t = 0; t < 8; ++t) acc[t] = (v8f)0.0f;

    for (int k0 = 0; k0 < ROWF; k0 += 48) {
        if (tid < 32) {
            // A tile: 64 rows x 48 dw, stride 432; pad 4dw/16dw -> stride 60
            tdm_load_2d(sA_off, XK + (size_t)row0 * ROWF + k0,
                        48, 48, 64, ROWF, /*pad_int=*/3, /*pad_amt=*/3);
            // W tile: contiguous 48*256 dw block (1D)
            tdm_load_2d(sW_off, Wp + (size_t)k0 * 256,
                        48 * 256, 48 * 256, 1, 48 * 256, 0, 0);
            __builtin_amdgcn_s_wait_tensorcnt(0);
        }
        __syncthreads();
#pragma unroll
        for (int kk = 0; kk < 48; kk += 4) {
            int kr = kk + lhalf * 2;
            v2f a = *(const v2f*)&sA[(rstrip + lmod) * SA_STRIDE + kk +
                                     ((kk >> 4) << 2) + lhalf * 2];
#pragma unroll
            for (int ct = 0; ct < 8; ++ct) {
                v2f bf;
                bf.x = sW[(kr + 0) * 256 + chalf + ct * 16 + lmod];
                bf.y = sW[(kr + 1) * 256 + chalf + ct * 16 + lmod];
                acc[ct] = wmma_f32(a, bf, acc[ct]);
            }
        }
        __syncthreads();
    }

#pragma unroll
    for (int ct = 0; ct < 8; ++ct) {
#pragma unroll
        for (int v = 0; v < 8; ++v) {
            int row = row0 + rstrip + lhalf * 8 + v;
            int j = chalf + ct * 16 + lmod;
            float s = 1.0f / (1.0f + __expf(-(acc[ct][v] + gb[j])));
            if (j < UDIM) {
                // r * h -> x0' slot (channel j+1 of section 0), in place
                XK[(size_t)row * ROWF + (j + 1)] = s * h[(size_t)row * UDIM + j];
            } else {
                Ubuf[(size_t)row * UDIM + (j - UDIM)] = s;
            }
        }
    }
}

// ---------------------------------------------------------------------------
// Candidate GEMM (64 x 128 per block, K=432) fused with GRU update and the
// 128->1 projection (shfl_xor lane reduce + LDS atomicAdd across col halves).
// ---------------------------------------------------------------------------
__global__ __launch_bounds__(256) void gemm_cand_kernel(
    const float* __restrict__ XK, const float* __restrict__ Wp,
    const float* __restrict__ cb, const float* __restrict__ h,
    const float* __restrict__ Ubuf, const float* __restrict__ pw,
    const float* __restrict__ pb, float* __restrict__ newh,
    float* __restrict__ out) {
    __shared__ float sA[64 * SA_STRIDE];
    __shared__ float sW[48 * 128];
    __shared__ float pbuf[64];

    int row0 = blockIdx.x * 64;
    int tid = threadIdx.x;
    int wave = tid >> 5, lane = tid & 31;
    int lhalf = lane >> 4, lmod = lane & 15;
    int rstrip = (wave & 3) * 16;
    int chalf  = (wave >> 2) * 64;
    unsigned sA_off = lds_offset_of(&sA[0]);
    unsigned sW_off = lds_offset_of(&sW[0]);

    if (tid < 64) pbuf[tid] = pb[0];

    v8f acc[4];
#pragma unroll
    for (int t = 0; t < 4; ++t) acc[t] = (v8f)0.0f;

    for (int k0 = 0; k0 < ROWF; k0 += 48) {
        if (tid < 32) {
            tdm_load_2d(sA_off, XK + (size_t)row0 * ROWF + k0,
                        48, 48, 64, ROWF, 3, 3);
            tdm_load_2d(sW_off, Wp + (size_t)k0 * 128,
                        48 * 128, 48 * 128, 1, 48 * 128, 0, 0);
            __builtin_amdgcn_s_wait_tensorcnt(0);
        }
        __syncthreads();
#pragma unroll
        for (int kk = 0; kk < 48; kk += 4) {
            int kr = kk + lhalf * 2;
            v2f a = *(const v2f*)&sA[(rstrip + lmod) * SA_STRIDE + kk +
                                     ((kk >> 4) << 2) + lhalf * 2];
#pragma unroll
            for (int ct = 0; ct < 4; ++ct) {
                v2f bf;
                bf.x = sW[(kr + 0) * 128 + chalf + ct * 16 + lmod];
                bf.y = sW[(kr + 1) * 128 + chalf + ct * 16 + lmod];
                acc[ct] = wmma_f32(a, bf, acc[ct]);
            }
        }
        __syncthreads();
    }

    float pacc[8];
#pragma unroll
    for (int v = 0; v < 8; ++v) pacc[v] = 0.0f;

#pragma unroll
    for (int ct = 0; ct < 4; ++ct) {
#pragma unroll
        for (int v = 0; v < 8; ++v) {
            int row = row0 + rstrip + lhalf * 8 + v;
            int j = chalf + ct * 16 + lmod;
            float c = tanhf(acc[ct][v] + cb[j]);
            float u = Ubuf[(size_t)row * UDIM + j];
            float hv = h[(size_t)row * UDIM + j];
            float nh = u * hv + (1.0f - u) * c;
            newh[(size_t)row * UDIM + j] = nh;
            pacc[v] += nh * pw[j];   // fused projection partial
        }
    }
    // Reduce projection partials across the 16 lanes of each half.
#pragma unroll
    for (int v = 0; v < 8; ++v) {
#pragma unroll
        for (int m = 1; m < 16; m <<= 1) pacc[v] += __shfl_xor(pacc[v], m, 32);
    }
    if (lmod == 0) {
#pragma unroll
        for (int v = 0; v < 8; ++v) {
            int lrow = rstrip + lhalf * 8 + v;
            atomicAdd(&pbuf[lrow], pacc[v]);   // combine the two column halves
        }
    }
    __syncthreads();
    if (tid < 64) out[row0 + tid] = pbuf[tid];
}

// ---------------------------------------------------------------------------
extern "C" void kernel_launch(void* const* d_in, const int* in_sizes, int n_in,
                              void* d_out, int out_size, void* d_ws, size_t ws_size,
                              hipStream_t stream) {
    const float* inputs = (const float*)d_in[0];
    const float* hidden = (const float*)d_in[1];   // (1,B,N*U)
    const float* support = (const float*)d_in[2];
    const float* gate_w = (const float*)d_in[3];
    const float* gate_b = (const float*)d_in[4];
    const float* cand_w = (const float*)d_in[5];
    const float* cand_b = (const float*)d_in[6];
    const float* proj_w = (const float*)d_in[7];
    const float* proj_b = (const float*)d_in[8];

    float* out = (float*)d_out;                    // (B, N*OD) = 131072
    float* newh = out + (size_t)NROWS;             // (1,B,N*U) = 16.7M

    float* ws = (float*)d_ws;
    float* XK   = ws;                                      // NROWS*432
    float* Ubuf = XK + (size_t)NROWS * ROWF;               // NROWS*128
    float* Wpg  = Ubuf + (size_t)NROWS * UDIM;             // 432*256
    float* Wpc  = Wpg + (size_t)ROWF * 256;                // 432*128

    prep_w_kernel<<<(ROWF * 256 + 255) / 256, 256, 0, stream>>>(gate_w, cand_w,
                                                                Wpg, Wpc);
    build_x0_kernel<<<(NROWS * CPAD + 255) / 256, 256, 0, stream>>>(XK, inputs,
                                                                    hidden);
    // Gate gconv: x1 = S x0 ; x2 = 2 S x1 - x0
    diffusion_kernel<<<BATCH * 16, 128, 0, stream>>>(support, XK, 0, 1, 0, 1.0f, 0.0f);
    diffusion_kernel<<<BATCH * 16, 128, 0, stream>>>(support, XK, 1, 2, 0, 2.0f, -1.0f);
    gemm_gate_kernel<<<NROWS / 64, 256, 0, stream>>>(XK, Wpg, gate_b, hidden, Ubuf);
    // Candidate gconv on x0' = concat(inputs, r*h) (written in place)
    diffusion_kernel<<<BATCH * 16, 128, 0, stream>>>(support, XK, 0, 1, 0, 1.0f, 0.0f);
    diffusion_kernel<<<BATCH * 16, 128, 0, stream>>>(support, XK, 1, 2, 0, 2.0f, -1.0f);
    gemm_cand_kernel<<<NROWS / 64, 256, 0, stream>>>(XK, Wpc, cand_b, hidden, Ubuf,
                                                     proj_w, proj_b, newh, out);
}